// Net_46007689674794
// MI455X (gfx1250) — compile-verified
//
#include <hip/hip_runtime.h>
#include <math.h>

#define N_NODES 50000
#define E_EDGES 1600000
#define R_REL   50
#define B_BASES 30
#define H_DIM   16
#define C_DIM   4
#define J_DIM   (N_NODES * H_DIM)   // 800000 columns of the w1 GEMM

#define M_PAD   64                  // R padded to 4 WMMA M-tiles
#define K_PAD   32                  // B padded to 8 WMMA K-steps

typedef __attribute__((ext_vector_type(2))) float v2f;
typedef __attribute__((ext_vector_type(8))) float v8f;

// Workspace layout (float offsets, all 16B aligned)
#define OFF_ACC1 0                       // N*H   = 800000
#define OFF_CNT  800000                  // N     = 50000
#define OFF_X    850000                  // N*H   = 800000
#define OFF_W2   1650000                 // R*H*C = 3200
#define OFF_OUT  1653200                 // N*C   = 200000
#define OFF_W1   1853200                 // R*N*H = 40000000
#define W1_FLOATS 40000000

// ---------------------------------------------------------------------------
// Kernel 1: w1[r, n, h] = sum_b comp1[r,b] * basis1[b, n, h]
// GEMM  C[64(pad of 50), 800000] = A[64,32(pad of 30)] x B[32,800000]
// One wave handles one 16-column tile, all 4 M-tiles, 8 K-steps of 4.
// A is zero-padded in LDS so the inner loop is branch-free (full EXEC):
//   2x global_load_b32 (B) + 4x ds_load_b64 (A) + 4x v_wmma_f32_16x16x4_f32.
// For k >= 30 the padded A rows are zero, so B's row index is merely clamped
// in-bounds (0 * finite = 0) instead of guarded.
// ---------------------------------------------------------------------------
__global__ __launch_bounds__(256) void w1_gemm_kernel(
    const float* __restrict__ comp1,   // [R, B] row-major
    const float* __restrict__ basis1,  // [B, J] row-major (b, n*H+h)
    float* __restrict__ w1)            // [R, J] row-major
{
    __shared__ float sA[M_PAD * K_PAD];  // 8 KB, zero-padded comp1 [m][k]
    for (int i = threadIdx.x; i < M_PAD * K_PAD; i += blockDim.x) {
        const int m = i >> 5, k = i & (K_PAD - 1);
        sA[i] = (m < R_REL && k < B_BASES) ? comp1[m * B_BASES + k] : 0.0f;
    }
    __syncthreads();

    const int lane = threadIdx.x & 31;
    const int wave = threadIdx.x >> 5;
    const int jt   = blockIdx.x * 8 + wave;       // 16-column tile index
    const int JT   = J_DIM / 16;                  // 50000
    if (jt >= JT) return;                         // wave-uniform exit

    const int col  = jt * 16 + (lane & 15);
    const int hi   = lane >> 4;                   // 0: lanes 0-15, 1: lanes 16-31
    const int mrow = lane & 15;

    v8f acc[4];
    #pragma unroll
    for (int mt = 0; mt < 4; ++mt) acc[mt] = (v8f){};

    #pragma unroll
    for (int kk = 0; kk < 8; ++kk) {
        const int k0 = kk * 4 + hi * 2;           // this lane's K pair base
        // Clamp row index in-bounds; A's zero padding kills k>=30 products.
        const int kx = (k0     < B_BASES) ? k0     : (B_BASES - 1);
        const int ky = (k0 + 1 < B_BASES) ? k0 + 1 : (B_BASES - 1);
        v2f b;
        b.x = basis1[(size_t)kx * J_DIM + col];
        b.y = basis1[(size_t)ky * J_DIM + col];

        #pragma unroll
        for (int mt = 0; mt < 4; ++mt) {
            const v2f a = *(const v2f*)&sA[(mt * 16 + mrow) * K_PAD + k0];
            acc[mt] = __builtin_amdgcn_wmma_f32_16x16x4_f32(
                false, a, false, b, (short)0, acc[mt], false, false);
        }
    }

    // Store: D element v in lane L is (M = v + 8*hi, N = lane&15) of the tile
    #pragma unroll
    for (int mt = 0; mt < 4; ++mt) {
        #pragma unroll
        for (int v = 0; v < 8; ++v) {
            const int row = mt * 16 + v + hi * 8;
            if (row < R_REL)
                w1[(size_t)row * J_DIM + col] = acc[mt][v];
        }
    }
}

// ---------------------------------------------------------------------------
// Kernel 2a: layer-1 edge scatter using materialized w1 (64B gather per edge)
// ---------------------------------------------------------------------------
__global__ __launch_bounds__(256) void l1_edges_w1_kernel(
    const int* __restrict__ edge_index, const int* __restrict__ edge_type,
    const float* __restrict__ w1, float* __restrict__ acc1, float* __restrict__ cnt)
{
    const int e = blockIdx.x * blockDim.x + threadIdx.x;
    if (e >= E_EDGES) return;
    const int src = edge_index[e];
    const int dst = edge_index[E_EDGES + e];
    const int r   = edge_type[e];

    const float4* wr = (const float4*)(w1 + ((size_t)r * N_NODES + src) * H_DIM);
    float4 m0 = wr[0], m1 = wr[1], m2 = wr[2], m3 = wr[3];

    float* out = acc1 + (size_t)dst * H_DIM;
    atomicAdd(out + 0,  m0.x); atomicAdd(out + 1,  m0.y);
    atomicAdd(out + 2,  m0.z); atomicAdd(out + 3,  m0.w);
    atomicAdd(out + 4,  m1.x); atomicAdd(out + 5,  m1.y);
    atomicAdd(out + 6,  m1.z); atomicAdd(out + 7,  m1.w);
    atomicAdd(out + 8,  m2.x); atomicAdd(out + 9,  m2.y);
    atomicAdd(out + 10, m2.z); atomicAdd(out + 11, m2.w);
    atomicAdd(out + 12, m3.x); atomicAdd(out + 13, m3.y);
    atomicAdd(out + 14, m3.z); atomicAdd(out + 15, m3.w);
    atomicAdd(cnt + dst, 1.0f);
}

// ---------------------------------------------------------------------------
// Kernel 2b: fallback (no w1 workspace): gather 30 basis rows per edge (L2)
// ---------------------------------------------------------------------------
__global__ __launch_bounds__(256) void l1_edges_direct_kernel(
    const int* __restrict__ edge_index, const int* __restrict__ edge_type,
    const float* __restrict__ comp1, const float* __restrict__ basis1,
    float* __restrict__ acc1, float* __restrict__ cnt)
{
    const int e = blockIdx.x * blockDim.x + threadIdx.x;
    if (e >= E_EDGES) return;
    const int src = edge_index[e];
    const int dst = edge_index[E_EDGES + e];
    const int r   = edge_type[e];

    float acc[H_DIM];
    #pragma unroll
    for (int h = 0; h < H_DIM; ++h) acc[h] = 0.0f;
    for (int b = 0; b < B_BASES; ++b) {
        const float cb = comp1[r * B_BASES + b];
        const float* br = basis1 + ((size_t)b * N_NODES + src) * H_DIM;
        #pragma unroll
        for (int h = 0; h < H_DIM; ++h) acc[h] += cb * br[h];
    }
    float* out = acc1 + (size_t)dst * H_DIM;
    #pragma unroll
    for (int h = 0; h < H_DIM; ++h) atomicAdd(out + h, acc[h]);
    atomicAdd(cnt + dst, 1.0f);
}

// ---------------------------------------------------------------------------
// Kernel 3: x = relu(acc1 / max(cnt,1) + root1 + bias1)
// ---------------------------------------------------------------------------
__global__ __launch_bounds__(256) void finalize1_kernel(
    const float* __restrict__ acc1, const float* __restrict__ cnt,
    const float* __restrict__ root1, const float* __restrict__ bias1,
    float* __restrict__ x)
{
    const int i = blockIdx.x * blockDim.x + threadIdx.x;
    if (i >= N_NODES * H_DIM) return;
    const int n = i >> 4, h = i & 15;
    const float c = fmaxf(cnt[n], 1.0f);
    const float v = acc1[i] / c + root1[i] + bias1[h];
    x[i] = fmaxf(v, 0.0f);
}

// ---------------------------------------------------------------------------
// Kernel 4: w2[r,h,c] = sum_b comp2[r,b]*basis2[b,h,c]   (tiny: 3200 outputs)
// ---------------------------------------------------------------------------
__global__ __launch_bounds__(256) void w2_kernel(
    const float* __restrict__ comp2, const float* __restrict__ basis2,
    float* __restrict__ w2)
{
    const int i = blockIdx.x * blockDim.x + threadIdx.x;
    if (i >= R_REL * H_DIM * C_DIM) return;
    const int r  = i / (H_DIM * C_DIM);
    const int hc = i % (H_DIM * C_DIM);
    float s = 0.0f;
    for (int b = 0; b < B_BASES; ++b)
        s += comp2[r * B_BASES + b] * basis2[b * H_DIM * C_DIM + hc];
    w2[i] = s;
}

// ---------------------------------------------------------------------------
// Kernel 5: layer-2 edge transform + scatter (w2 cached in LDS, 12.8 KB)
// ---------------------------------------------------------------------------
__global__ __launch_bounds__(256) void l2_edges_kernel(
    const int* __restrict__ edge_index, const int* __restrict__ edge_type,
    const float* __restrict__ x, const float* __restrict__ w2,
    float* __restrict__ out_acc)
{
    __shared__ float sw2[R_REL * H_DIM * C_DIM];
    for (int i = threadIdx.x; i < R_REL * H_DIM * C_DIM; i += blockDim.x)
        sw2[i] = w2[i];
    __syncthreads();

    const int e = blockIdx.x * blockDim.x + threadIdx.x;
    if (e >= E_EDGES) return;
    const int src = edge_index[e];
    const int dst = edge_index[E_EDGES + e];
    const int r   = edge_type[e];

    const float4* xr = (const float4*)(x + (size_t)src * H_DIM);
    float4 x0 = xr[0], x1 = xr[1], x2 = xr[2], x3 = xr[3];
    const float xv[H_DIM] = {x0.x, x0.y, x0.z, x0.w, x1.x, x1.y, x1.z, x1.w,
                             x2.x, x2.y, x2.z, x2.w, x3.x, x3.y, x3.z, x3.w};
    const float* w = sw2 + r * H_DIM * C_DIM;   // [h][c]
    float m[C_DIM] = {0.0f, 0.0f, 0.0f, 0.0f};
    #pragma unroll
    for (int h = 0; h < H_DIM; ++h) {
        const float xh = xv[h];
        m[0] += xh * w[h * 4 + 0];
        m[1] += xh * w[h * 4 + 1];
        m[2] += xh * w[h * 4 + 2];
        m[3] += xh * w[h * 4 + 3];
    }
    float* out = out_acc + (size_t)dst * C_DIM;
    atomicAdd(out + 0, m[0]); atomicAdd(out + 1, m[1]);
    atomicAdd(out + 2, m[2]); atomicAdd(out + 3, m[3]);
}

// ---------------------------------------------------------------------------
// Kernel 6: out = log_softmax(out_acc/max(cnt,1) + x@root2 + bias2)
// ---------------------------------------------------------------------------
__global__ __launch_bounds__(256) void finalize2_kernel(
    const float* __restrict__ out_acc, const float* __restrict__ cnt,
    const float* __restrict__ x, const float* __restrict__ root2,
    const float* __restrict__ bias2, float* __restrict__ out)
{
    const int n = blockIdx.x * blockDim.x + threadIdx.x;
    if (n >= N_NODES) return;
    const float c = fmaxf(cnt[n], 1.0f);
    const float* xr = x + (size_t)n * H_DIM;

    float v[C_DIM];
    #pragma unroll
    for (int cc = 0; cc < C_DIM; ++cc) {
        float s = out_acc[(size_t)n * C_DIM + cc] / c + bias2[cc];
        #pragma unroll
        for (int h = 0; h < H_DIM; ++h) s += xr[h] * root2[h * C_DIM + cc];
        v[cc] = s;
    }
    const float mx  = fmaxf(fmaxf(v[0], v[1]), fmaxf(v[2], v[3]));
    const float sum = expf(v[0] - mx) + expf(v[1] - mx) +
                      expf(v[2] - mx) + expf(v[3] - mx);
    const float lse = logf(sum);
    #pragma unroll
    for (int cc = 0; cc < C_DIM; ++cc)
        out[(size_t)n * C_DIM + cc] = v[cc] - mx - lse;
}

extern "C" void kernel_launch(void* const* d_in, const int* in_sizes, int n_in,
                              void* d_out, int out_size, void* d_ws, size_t ws_size,
                              hipStream_t stream) {
    const int*   edge_index = (const int*)d_in[0];
    const int*   edge_type  = (const int*)d_in[1];
    // d_in[2] = edge_norm (unused by the reference forward)
    const float* basis1 = (const float*)d_in[3];
    const float* comp1  = (const float*)d_in[4];
    const float* root1  = (const float*)d_in[5];
    const float* bias1  = (const float*)d_in[6];
    const float* basis2 = (const float*)d_in[7];
    const float* comp2  = (const float*)d_in[8];
    const float* root2  = (const float*)d_in[9];
    const float* bias2  = (const float*)d_in[10];

    float* ws      = (float*)d_ws;
    float* acc1    = ws + OFF_ACC1;
    float* cnt     = ws + OFF_CNT;
    float* xbuf    = ws + OFF_X;
    float* w2buf   = ws + OFF_W2;
    float* out_acc = ws + OFF_OUT;
    float* w1      = ws + OFF_W1;

    const bool use_w1 =
        ws_size >= (size_t)(OFF_W1 + W1_FLOATS) * sizeof(float);

    // Zero the atomic accumulators (acc1+cnt are contiguous) every call.
    hipMemsetAsync(acc1, 0, (size_t)(N_NODES * H_DIM + N_NODES) * sizeof(float), stream);
    hipMemsetAsync(out_acc, 0, (size_t)(N_NODES * C_DIM) * sizeof(float), stream);

    const int eb = (E_EDGES + 255) / 256;

    if (use_w1) {
        // WMMA GEMM: 50000 column tiles, 8 waves per 256-thread block
        w1_gemm_kernel<<<(J_DIM / 16 + 7) / 8, 256, 0, stream>>>(comp1, basis1, w1);
        l1_edges_w1_kernel<<<eb, 256, 0, stream>>>(edge_index, edge_type, w1, acc1, cnt);
    } else {
        l1_edges_direct_kernel<<<eb, 256, 0, stream>>>(edge_index, edge_type,
                                                       comp1, basis1, acc1, cnt);
    }

    finalize1_kernel<<<(N_NODES * H_DIM + 255) / 256, 256, 0, stream>>>(
        acc1, cnt, root1, bias1, xbuf);
    w2_kernel<<<(R_REL * H_DIM * C_DIM + 255) / 256, 256, 0, stream>>>(
        comp2, basis2, w2buf);
    l2_edges_kernel<<<eb, 256, 0, stream>>>(edge_index, edge_type, xbuf, w2buf, out_acc);
    finalize2_kernel<<<(N_NODES + 255) / 256, 256, 0, stream>>>(
        out_acc, cnt, xbuf, root2, bias2, (float*)d_out);
}